// HierarchicalGraphPooling_64965675319804
// MI455X (gfx1250) — compile-verified
//
#include <hip/hip_runtime.h>
#include <hip/hip_bf16.h>

// Problem constants (match reference)
#define BB 4
#define NN 2048
#define CC 256
#define EE 4
#define KK 1024          // k = N * 0.5
#define H1_ 64
#define H2_ 16

typedef __attribute__((ext_vector_type(2))) float v2f;
typedef __attribute__((ext_vector_type(8))) float v8f;

// ---------------------------------------------------------------------------
// Stage 1: score MLP.  Layer 1 (256 -> 64) via V_WMMA_F32_16X16X4_F32,
// layers 2/3 scalar (tiny).  One wave handles a 16-node tile.
// grid = 64 blocks x 256 threads (8 waves) -> 512 waves x 16 nodes = 8192.
// ---------------------------------------------------------------------------
__global__ __launch_bounds__(256)
void hgp_score_kernel(const float* __restrict__ x,
                      const float* __restrict__ W1, const float* __restrict__ b1,
                      const float* __restrict__ W2, const float* __restrict__ b2,
                      const float* __restrict__ W3, const float* __restrict__ b3,
                      float* __restrict__ scores) {
    __shared__ float h1s[8][16][H1_];   // 32 KB

    const int wave = threadIdx.x >> 5;
    const int lane = threadIdx.x & 31;
    const int hlf  = lane >> 4;         // 0: lanes 0-15, 1: lanes 16-31
    const int lrow = lane & 15;

    const int tile = blockIdx.x * 8 + wave;        // 16-node tile id (0..511)
    const int node = tile * 16 + lrow;             // flat node index in [0, B*N)
    const float* xrow = x + (size_t)node * CC;     // A-matrix row (M = lrow)

    // 4 N-tiles of 16 columns each cover H1 = 64
    for (int nt = 0; nt < 4; ++nt) {
        const int n = nt * 16 + lrow;              // B-matrix column (N = lrow)
        v8f acc = {};
        for (int k = 0; k < CC; k += 4) {
            const int kk = k + 2 * hlf;            // this lane's K pair
            v2f a, bfr;
            a.x   = xrow[kk];
            a.y   = xrow[kk + 1];
            bfr.x = W1[(size_t)kk * H1_ + n];
            bfr.y = W1[(size_t)(kk + 1) * H1_ + n];
            acc = __builtin_amdgcn_wmma_f32_16x16x4_f32(
                false, a, false, bfr, (short)0, acc, false, false);
        }
        // C/D layout: VGPR r -> M = r + 8*hlf, N = lrow
        const float bias = b1[n];
#pragma unroll
        for (int r = 0; r < 8; ++r) {
            float v = acc[r] + bias;
            h1s[wave][r + 8 * hlf][n] = v > 0.0f ? v : 0.0f;
        }
    }
    __syncthreads();

    // Layers 2 & 3: lanes 0-15 each finish one node (tiny: ~1.1k FMA).
    if (hlf == 0) {
        const float* h1 = h1s[wave][lrow];
        float h2[H2_];
#pragma unroll
        for (int j = 0; j < H2_; ++j) {
            float s = b2[j];
            for (int h = 0; h < H1_; ++h)
                s += h1[h] * W2[h * H2_ + j];
            h2[j] = s > 0.0f ? s : 0.0f;
        }
        float sc = b3[0];
#pragma unroll
        for (int j = 0; j < H2_; ++j)
            sc += h2[j] * W3[j];
        scores[node] = sc;
    }
}

// ---------------------------------------------------------------------------
// Stage 2: exact top-k via full bitonic sort of 2048 keys per batch.
// Key = (descending-monotone(score) << 32) | index  -> ascending sort
// reproduces jax.lax.top_k order (descending score, ascending index on ties).
// grid = B blocks x 1024 threads, 16 KB LDS.
// ---------------------------------------------------------------------------
__global__ __launch_bounds__(1024)
void hgp_topk_kernel(const float* __restrict__ scores, int* __restrict__ idx) {
    __shared__ unsigned long long keys[NN];
    const int b   = blockIdx.x;
    const int tid = threadIdx.x;

    for (int i = tid; i < NN; i += 1024) {
        unsigned u = __float_as_uint(scores[b * NN + i]);
        u = (u & 0x80000000u) ? ~u : (u | 0x80000000u);   // ascending-monotone
        unsigned d = ~u;                                   // descending
        keys[i] = ((unsigned long long)d << 32) | (unsigned)i;
    }
    __syncthreads();

    for (int k = 2; k <= NN; k <<= 1) {
        for (int j = k >> 1; j > 0; j >>= 1) {
            for (int t = tid; t < NN; t += 1024) {
                const int ixj = t ^ j;
                if (ixj > t) {
                    const bool up = ((t & k) == 0);
                    unsigned long long a = keys[t], c = keys[ixj];
                    if ((a > c) == up) { keys[t] = c; keys[ixj] = a; }
                }
            }
            __syncthreads();
        }
    }
    if (tid < KK)
        idx[b * KK + tid] = (int)(keys[tid] & 0xFFFFFFFFu);
}

// ---------------------------------------------------------------------------
// Stage 3a: pooled_features[b,i,:] = x[b, idx[b,i], :]   (float4 coalesced)
// one float4 per thread; grid covers B*K*C/4 elements.
// ---------------------------------------------------------------------------
__global__ __launch_bounds__(256)
void hgp_gather_feat(const float* __restrict__ x, const int* __restrict__ idx,
                     float* __restrict__ out) {
    const int g   = blockIdx.x * 256 + threadIdx.x;   // float4 id
    const int row = g >> 6;                            // b*K + i  (C/4 = 64)
    const int c4  = g & 63;
    const int b   = row >> 10;
    const int r   = idx[row];
    const float4* src = (const float4*)(x + ((size_t)(b * NN + r)) * CC);
    float4*       dst = (float4*)(out + (size_t)row * CC);
    dst[c4] = src[c4];
}

// ---------------------------------------------------------------------------
// Stage 3b: pooled_adjacency[b,i,j] = adj[b, idx[b,i], idx[b,j]]
// one block per output row; column indices staged in LDS; the 8 KB source
// row stays cache-resident while its 1024 random columns are picked.
// ---------------------------------------------------------------------------
__global__ __launch_bounds__(256)
void hgp_gather_adj(const float* __restrict__ adj, const int* __restrict__ idx,
                    float* __restrict__ out) {
    __shared__ int jd[KK];
    const int row = blockIdx.x;         // b*K + i
    const int b   = row >> 10;
    for (int t = threadIdx.x; t < KK; t += 256)
        jd[t] = idx[(b << 10) + t];
    __syncthreads();
    const int r = idx[row];
    const float* src = adj + ((size_t)b * NN + r) * NN;
    float*       dst = out + (size_t)row * KK;
    for (int t = threadIdx.x; t < KK; t += 256)
        dst[t] = src[jd[t]];
}

// ---------------------------------------------------------------------------
// Stage 3c: pooled_edge_features[b,i,:,:] = edge[b, idx[b,i], :, :]
// pure 32 KB contiguous row copy -> float4 streaming (HBM-bound: 256 MB).
// ---------------------------------------------------------------------------
__global__ __launch_bounds__(256)
void hgp_gather_edge(const float* __restrict__ ef, const int* __restrict__ idx,
                     float* __restrict__ out) {
    const int row = blockIdx.x;         // b*K + i
    const int b   = row >> 10;
    const int r   = idx[row];
    const float4* src = (const float4*)(ef + ((size_t)b * NN + r) * NN * EE);
    float4*       dst = (float4*)(out + (size_t)row * NN * EE);
    for (int t = threadIdx.x; t < NN * EE / 4; t += 256)
        dst[t] = src[t];
}

extern "C" void kernel_launch(void* const* d_in, const int* in_sizes, int n_in,
                              void* d_out, int out_size, void* d_ws, size_t ws_size,
                              hipStream_t stream) {
    const float* x   = (const float*)d_in[0];
    const float* adj = (const float*)d_in[1];
    const float* ef  = (const float*)d_in[2];
    // d_in[3] = superpoint_centroids (unused by reference outputs)
    const float* W1 = (const float*)d_in[4];
    const float* b1 = (const float*)d_in[5];
    const float* W2 = (const float*)d_in[6];
    const float* b2 = (const float*)d_in[7];
    const float* W3 = (const float*)d_in[8];
    const float* b3 = (const float*)d_in[9];

    float* scores = (float*)d_ws;                              // B*N floats
    int*   idx    = (int*)((char*)d_ws + sizeof(float) * BB * NN);  // B*K ints

    float* out_feat = (float*)d_out;                   // [B,K,C]
    float* out_adj  = out_feat + (size_t)BB * KK * CC; // [B,K,K]
    float* out_edge = out_adj + (size_t)BB * KK * KK;  // [B,K,N,E]

    hgp_score_kernel<<<64, 256, 0, stream>>>(x, W1, b1, W2, b2, W3, b3, scores);
    hgp_topk_kernel<<<BB, 1024, 0, stream>>>(scores, idx);
    hgp_gather_feat<<<(BB * KK * CC / 4) / 256, 256, 0, stream>>>(x, idx, out_feat);
    hgp_gather_adj<<<BB * KK, 256, 0, stream>>>(adj, idx, out_adj);
    hgp_gather_edge<<<BB * KK, 256, 0, stream>>>(ef, idx, out_edge);
}